// TFAdaptiveAvgPool1D_40003325395108
// MI455X (gfx1250) — compile-verified
//
#include <hip/hip_runtime.h>

// out[j] = (1/W) * sum_{k<W} in[W*j + k]   with W = in_dim/out_dim = 8.
// Memory-bound (604 MB @ 23.3 TB/s ~ 26 us). CDNA5 path: double-buffered
// global_load_async_to_lds_b128 (ASYNCcnt) staging + ds_load_b128 readback.

#define ASYNC_B128(LDSOFF, GPTR, OFF)                                          \
  asm volatile("global_load_async_to_lds_b128 %0, %1, off offset:" OFF         \
               :: "v"(LDSOFF), "v"(GPTR) : "memory")

namespace {
constexpr int      THREADS   = 256;
constexpr int      OPT       = THREADS * 4;     // outputs per tile (1024)
constexpr int      IPT       = OPT * 8;         // input floats per tile (8192 = 32 KiB)
constexpr int      NT        = 8;               // tiles per block
constexpr unsigned TSTRIDE   = 144;             // LDS bytes/thread (128 data + 16 pad, kills bank conflicts)
constexpr unsigned TILE_LDS  = THREADS * TSTRIDE; // 36864 B per buffer
}

__global__ __launch_bounds__(THREADS)
void pool8_async_kernel(const float* __restrict__ in, float* __restrict__ out) {
  __shared__ __align__(16) unsigned char smem[2 * TILE_LDS];   // 72 KiB

  const unsigned tid   = threadIdx.x;
  const unsigned lbase = (unsigned)(size_t)(&smem[0]) + tid * TSTRIDE;
  const long     tile0 = (long)blockIdx.x * NT;

  auto issue_tile = [&](int t, int buf) {
    // This thread DMAs its own 128 contiguous bytes of input for tile t.
    const char*    g = (const char*)(in + (tile0 + t) * (long)IPT) + (long)tid * 128;
    const unsigned l = lbase + (unsigned)buf * TILE_LDS;
    ASYNC_B128(l, g, "0");
    ASYNC_B128(l, g, "16");
    ASYNC_B128(l, g, "32");
    ASYNC_B128(l, g, "48");
    ASYNC_B128(l, g, "64");
    ASYNC_B128(l, g, "80");
    ASYNC_B128(l, g, "96");
    ASYNC_B128(l, g, "112");
  };

  issue_tile(0, 0);                              // prologue: tile 0 -> buf 0

  for (int t = 0; t < NT; ++t) {
    if (t + 1 < NT) {
      issue_tile(t + 1, (t + 1) & 1);            // prefetch next tile into other buffer
      // 8 newer ops outstanding; ASYNCcnt completes in order, so <=8 means tile t landed.
      asm volatile("s_wait_asynccnt 0x8" ::: "memory");
    } else {
      asm volatile("s_wait_asynccnt 0x0" ::: "memory");
    }

    const float4* p = (const float4*)(smem + (unsigned)(t & 1) * TILE_LDS + tid * TSTRIDE);
    float4 a0 = p[0], b0 = p[1];
    float4 a1 = p[2], b1 = p[3];
    float4 a2 = p[4], b2 = p[5];
    float4 a3 = p[6], b3 = p[7];

    float4 r;
    r.x = 0.125f * (((a0.x + a0.y) + (a0.z + a0.w)) + ((b0.x + b0.y) + (b0.z + b0.w)));
    r.y = 0.125f * (((a1.x + a1.y) + (a1.z + a1.w)) + ((b1.x + b1.y) + (b1.z + b1.w)));
    r.z = 0.125f * (((a2.x + a2.y) + (a2.z + a2.w)) + ((b2.x + b2.y) + (b2.z + b2.w)));
    r.w = 0.125f * (((a3.x + a3.y) + (a3.z + a3.w)) + ((b3.x + b3.y) + (b3.z + b3.w)));

    float4* o = (float4*)(out + (tile0 + t) * (long)OPT) + tid;
    *o = r;
  }
}

// Generic / remainder path: arbitrary uniform window W, one output per thread, guarded.
__global__ void pool_generic_kernel(const float* __restrict__ in, float* __restrict__ out,
                                    long start, long n_out, int w, float scale) {
  long j = start + (long)blockIdx.x * blockDim.x + threadIdx.x;
  if (j < n_out) {
    const float* p = in + j * (long)w;
    float s = 0.0f;
    for (int k = 0; k < w; ++k) s += p[k];
    out[j] = scale * s;
  }
}

extern "C" void kernel_launch(void* const* d_in, const int* in_sizes, int n_in,
                              void* d_out, int out_size, void* d_ws, size_t ws_size,
                              hipStream_t stream) {
  (void)d_ws; (void)ws_size; (void)n_in;
  const float* in  = (const float*)d_in[0];   // [B*C, in_dim] fp32
  float*       out = (float*)d_out;           // [B*C, out_dim] fp32
  // d_in[1] (pool_map) is block-diagonal uniform 1/W when in_dim % out_dim == 0.

  const long n_out = (long)out_size;
  const long n_in0 = (long)in_sizes[0];
  const long ratio = (n_out > 0) ? (n_in0 / n_out) : 0;

  if (ratio == 8 && n_in0 == 8 * n_out) {
    const long tiles  = n_out / OPT;
    const long blocks = tiles / NT;
    if (blocks > 0)
      pool8_async_kernel<<<(int)blocks, THREADS, 0, stream>>>(in, out);
    const long done = blocks * (long)NT * OPT;
    const long rem  = n_out - done;
    if (rem > 0) {
      int tb = (int)((rem + 255) / 256);
      pool_generic_kernel<<<tb, 256, 0, stream>>>(in, out, done, n_out, 8, 0.125f);
    }
  } else if (ratio > 0 && n_in0 == ratio * n_out) {
    int tb = (int)((n_out + 255) / 256);
    pool_generic_kernel<<<tb, 256, 0, stream>>>(in, out, 0, n_out, (int)ratio,
                                                1.0f / (float)ratio);
  }
}